// ChamferLoss_30056181137908
// MI455X (gfx1250) — compile-verified
//
#include <hip/hip_runtime.h>
#include <math.h>

// Chamfer loss on MI455X (gfx1250, wave32).
// dist_sq(n,m) = |p3|^2 + |q3|^2 - 2 p3.q3 per 16x16 tile with ONE
// V_WMMA_F32_16X16X4_F32:
//   A = (p1,p2,p3,|p|^2), B = (-2q1,-2q2,-2q3,1), C preloaded = splat(|q|^2)
// All min reductions run in the SIGNED-INT domain (bit pattern order ==
// float order for >=0; rounding-negatives clamp to 0 either way), which
// avoids the v_max_num_f32 x,x,x NaN-canonicalization the float path emits.
// sqrt applied once per point at the end (sqrt is monotone).
// Batch split across SPLIT workgroups by row tiles to fill the WGPs;
// q-side mins: LDS ds_min_i32 during the loop, one global_atomic_min_i32
// flush per entry, reduced by kernel 2.

typedef __attribute__((ext_vector_type(2))) float v2f;
typedef __attribute__((ext_vector_type(8))) float v8f;

#define NPTS 1024
#define NTILES 64          // 1024 / 16
#define CHAMFER_EPS 1e-16f
#define POS_INF_BITS 0x7F800000

__device__ __forceinline__ int imin(int a, int b) { return a < b ? a : b; }

template <int SPLIT>
__global__ __launch_bounds__(256) void chamfer_main(
    const float* __restrict__ p,   // [B,1024,4]
    const float* __restrict__ q,   // [B,1024,4]
    float* __restrict__ out,       // scalar
    int* __restrict__ qws)         // [B,1024] global q-side min (SPLIT>1)
{
    constexpr int TPW    = NTILES / SPLIT;  // row tiles per workgroup
    constexpr int KITERS = TPW / 8;         // row tiles per wave

    __shared__ float s_pv[NPTS * 4];        // (p1, p2, p3, |p|^2)
    __shared__ float s_bv[NPTS * 4];        // (-2q1, -2q2, -2q3, 1)
    __shared__ float s_qn[NPTS];            // |q|^2
    __shared__ int   s_qmin[NPTS];          // per-q min dist_sq bits (int order)

    const int b    = blockIdx.x / SPLIT;
    const int s    = blockIdx.x % SPLIT;
    const int tid  = threadIdx.x;
    const int lane = tid & 31;
    const int wave = tid >> 5;

    const float* pb = p + (size_t)b * NPTS * 4;
    const float* qb = q + (size_t)b * NPTS * 4;

    // ---- Stage + transform into LDS (coalesced float4 loads) ----
    for (int i = tid; i < NPTS; i += 256) {
        float4 pt = ((const float4*)pb)[i];
        float  pn = pt.y * pt.y + pt.z * pt.z + pt.w * pt.w;
        s_pv[i * 4 + 0] = pt.y;
        s_pv[i * 4 + 1] = pt.z;
        s_pv[i * 4 + 2] = pt.w;
        s_pv[i * 4 + 3] = pn;

        float4 qt = ((const float4*)qb)[i];
        float  qn = qt.y * qt.y + qt.z * qt.z + qt.w * qt.w;
        s_bv[i * 4 + 0] = -2.0f * qt.y;
        s_bv[i * 4 + 1] = -2.0f * qt.z;
        s_bv[i * 4 + 2] = -2.0f * qt.w;
        s_bv[i * 4 + 3] = 1.0f;
        s_qn[i]   = qn;
        s_qmin[i] = POS_INF_BITS;
    }
    __syncthreads();

    // Fragment addressing (f32 K=4 layout: lanes 0-15 -> K=0,1 ;
    // lanes 16-31 -> K=2,3 ; row/col = lane % 16).
    const int col0 = lane & 15;
    const int kb   = (lane < 16) ? 0 : 2;

    // Preload this wave's A fragments (row tiles: s*TPW + wave + 8*k)
    v2f a1[KITERS];
    #pragma unroll
    for (int k = 0; k < KITERS; ++k) {
        const int row = (s * TPW + wave + 8 * k) * 16 + col0;
        a1[k].x = s_pv[row * 4 + kb];
        a1[k].y = s_pv[row * 4 + kb + 1];
    }

    int rowmin[KITERS][8];
    #pragma unroll
    for (int k = 0; k < KITERS; ++k)
        #pragma unroll
        for (int r = 0; r < 8; ++r) rowmin[k][r] = POS_INF_BITS;

    // ---- Main loop: column tiles outer, this wave's row tiles inner ----
    for (int ct = 0; ct < NTILES; ++ct) {
        const int col = ct * 16 + col0;
        v2f b1;
        b1.x = s_bv[col * 4 + kb];
        b1.y = s_bv[col * 4 + kb + 1];
        const float qv = s_qn[col];          // == qn[N] broadcast in C layout

        int cmin = POS_INF_BITS;
        #pragma unroll
        for (int k = 0; k < KITERS; ++k) {
            v8f c;
            #pragma unroll
            for (int r = 0; r < 8; ++r) c[r] = qv;   // C preloaded with |q|^2
            c = __builtin_amdgcn_wmma_f32_16x16x4_f32(
                    false, a1[k], false, b1, (short)0, c, false, false);
            // c[r]: lanes 0-15 -> dist_sq(rowtile*16+r,   col)
            //       lanes16-31 -> dist_sq(rowtile*16+r+8, col)
            int ci[8];
            #pragma unroll
            for (int r = 0; r < 8; ++r) {
                ci[r] = __float_as_int(c[r]);
                rowmin[k][r] = imin(rowmin[k][r], ci[r]);   // v_min_i32
            }
            int t = imin(imin(imin(ci[0], ci[1]), imin(ci[2], ci[3])),
                         imin(imin(ci[4], ci[5]), imin(ci[6], ci[7])));
            cmin = imin(cmin, t);                           // v_min3_i32 fodder
        }
        // all 32 lanes: lane and lane^16 share an address -> LDS combines halves
        atomicMin(&s_qmin[col], cmin);       // ds_min_i32
    }

    // ---- p-side: reduce row mins across the 16 column classes ----
    float rsum = 0.0f;
    #pragma unroll
    for (int k = 0; k < KITERS; ++k) {
        #pragma unroll
        for (int r = 0; r < 8; ++r) {
            int m = rowmin[k][r];
            m = imin(m, __shfl_xor(m, 1, 32));
            m = imin(m, __shfl_xor(m, 2, 32));
            m = imin(m, __shfl_xor(m, 4, 32));
            m = imin(m, __shfl_xor(m, 8, 32));
            float mf = fmaxf(__int_as_float(m), 0.0f);
            rsum += __builtin_amdgcn_sqrtf(mf + CHAMFER_EPS);
        }
    }
    rsum += __shfl_xor(rsum, 16, 32);        // halves held rows +0..7 / +8..15
    if (lane == 0) atomicAdd(out, 0.5f * rsum);

    __syncthreads();

    // ---- q-side ----
    if (SPLIT > 1) {
        // flush LDS mins to per-batch global min array (reduced by kernel 2)
        for (int i = tid; i < NPTS; i += 256)
            atomicMin(&qws[(size_t)b * NPTS + i], s_qmin[i]);
    } else {
        float qsum = 0.0f;
        for (int i = tid; i < NPTS; i += 256) {
            float m = fmaxf(__int_as_float(s_qmin[i]), 0.0f);
            qsum += __builtin_amdgcn_sqrtf(m + CHAMFER_EPS);
        }
        qsum += __shfl_xor(qsum, 1, 32);
        qsum += __shfl_xor(qsum, 2, 32);
        qsum += __shfl_xor(qsum, 4, 32);
        qsum += __shfl_xor(qsum, 8, 32);
        qsum += __shfl_xor(qsum, 16, 32);
        if (lane == 0) atomicAdd(out, 0.5f * qsum);
    }
}

__global__ __launch_bounds__(256) void chamfer_qreduce(
    const int* __restrict__ qws, float* __restrict__ out)
{
    const int b    = blockIdx.x;
    const int tid  = threadIdx.x;
    const int lane = tid & 31;

    float qsum = 0.0f;
    for (int i = tid; i < NPTS; i += 256) {
        float m = fmaxf(__int_as_float(qws[(size_t)b * NPTS + i]), 0.0f);
        qsum += __builtin_amdgcn_sqrtf(m + CHAMFER_EPS);
    }
    qsum += __shfl_xor(qsum, 1, 32);
    qsum += __shfl_xor(qsum, 2, 32);
    qsum += __shfl_xor(qsum, 4, 32);
    qsum += __shfl_xor(qsum, 8, 32);
    qsum += __shfl_xor(qsum, 16, 32);
    if (lane == 0) atomicAdd(out, 0.5f * qsum);
}

extern "C" void kernel_launch(void* const* d_in, const int* in_sizes, int n_in,
                              void* d_out, int out_size, void* d_ws, size_t ws_size,
                              hipStream_t stream) {
    const float* p = (const float*)d_in[0];
    const float* q = (const float*)d_in[1];
    float* out = (float*)d_out;

    const int B = in_sizes[0] / (NPTS * 4);   // 128 for the reference shapes
    const size_t qws_bytes = (size_t)B * NPTS * sizeof(int);

    hipMemsetAsync(out, 0, sizeof(float), stream);

    if (ws_size >= qws_bytes) {
        // split each batch over 4 WGs (by row tiles) to fill the WGPs;
        // redundant p/q staging traffic stays resident in the 192 MB L2.
        int* qws = (int*)d_ws;
        // 0x7F7F7F7F: large positive int AND large positive float bits
        hipMemsetAsync(qws, 0x7F, qws_bytes, stream);
        chamfer_main<4><<<B * 4, 256, 0, stream>>>(p, q, out, qws);
        chamfer_qreduce<<<B, 256, 0, stream>>>(qws, out);
    } else {
        chamfer_main<1><<<B, 256, 0, stream>>>(p, q, out, nullptr);
    }
}